// CriterionPairWiseforWholeFeatAfterPoolFeatureMaps_85057532330632
// MI455X (gfx1250) — compile-verified
//
#include <hip/hip_runtime.h>

typedef __attribute__((ext_vector_type(2))) float v2f;
typedef __attribute__((ext_vector_type(8))) float v8f;

// ---- problem constants (from reference: shape (2,4,32,64,64)) ----
constexpr int NMAT  = 8;          // B*L
constexpr int C     = 32;         // channels
constexpr int HW    = 4096;       // 64*64 spatial
constexpr int SPLIT = 16;         // M-splits per map
constexpr int COLS  = HW / SPLIT; // 256 columns per block
constexpr int TPB   = 128;        // 4 waves (wave32)
constexpr int WAVES = 4;
constexpr int TILES = 10;         // SS00,SS01,SS11, TT00,TT01,TT11, ST00,ST01,ST10,ST11
constexpr int TILE_ELEMS = 256;   // 16x16
constexpr int GRAM_FLOATS = TILES * TILE_ELEMS; // 2560 per partial

static __device__ __forceinline__ v8f wmma4(v2f a, v2f b, v8f c) {
    // D = A(16x4 f32) * B(4x16 f32) + C(16x16 f32); exact fp32 matrix op
    return __builtin_amdgcn_wmma_f32_16x16x4_f32(
        /*neg_a=*/false, a, /*neg_b=*/false, b,
        /*c_mod=*/(short)0, c, /*reuse_a=*/false, /*reuse_b=*/false);
}

// Kernel 1: per (map n, M-split s): column norms + 32x32 partial Grams via WMMA.
__global__ void __launch_bounds__(TPB)
gram_partial_kernel(const float* __restrict__ S, const float* __restrict__ T,
                    float* __restrict__ wsGram) {
    const int s = blockIdx.x;   // M split
    const int n = blockIdx.y;   // map index
    const float* Sn = S + (size_t)n * C * HW;
    const float* Tn = T + (size_t)n * C * HW;
    const int mbase = s * COLS;
    const int t = threadIdx.x;

    __shared__ float invS[COLS];
    __shared__ float invT[COLS];
    __shared__ float gram[WAVES][GRAM_FLOATS];

    // ---- phase A: per-column inverse norms (coalesced over m) ----
    for (int i = t; i < COLS; i += TPB) {
        const int m = mbase + i;
        float ss = 0.f, tt = 0.f;
        #pragma unroll
        for (int c = 0; c < C; ++c) {
            const float vs = Sn[c * HW + m];
            const float vt = Tn[c * HW + m];
            ss += vs * vs;
            tt += vt * vt;
        }
        invS[i] = 1.0f / (__builtin_sqrtf(ss) + 1e-8f);
        invT[i] = 1.0f / (__builtin_sqrtf(tt) + 1e-8f);
    }
    __syncthreads();

    // ---- phase B: WMMA accumulation over this wave's 64 columns ----
    const int wave = t >> 5;
    const int lane = t & 31;
    const int half = lane >> 4;   // K-half selector (A/B layout: lanes 16-31 hold K=2,3)
    const int cl   = lane & 15;   // channel-within-half / tile row/col

    v8f acc[TILES];
    #pragma unroll
    for (int k = 0; k < TILES; ++k) acc[k] = (v8f){0,0,0,0,0,0,0,0};

    const int chunks = COLS / WAVES / 4;  // 16 chunks of K=4
    for (int ch = 0; ch < chunks; ++ch) {
        const int cloc = wave * (COLS / WAVES) + ch * 4 + 2 * half; // even
        const int col0 = mbase + cloc;

        // lane holds rows (cl) and (cl+16), K = {2*half, 2*half+1} -> 2 consecutive m's
        const float2 rs0 = *reinterpret_cast<const float2*>(Sn + cl * HW + col0);
        const float2 rs1 = *reinterpret_cast<const float2*>(Sn + (cl + 16) * HW + col0);
        const float2 rt0 = *reinterpret_cast<const float2*>(Tn + cl * HW + col0);
        const float2 rt1 = *reinterpret_cast<const float2*>(Tn + (cl + 16) * HW + col0);

        const float wS0 = invS[cloc], wS1 = invS[cloc + 1];
        const float wT0 = invT[cloc], wT1 = invT[cloc + 1];

        const v2f s0 = {rs0.x * wS0, rs0.y * wS1};
        const v2f s1 = {rs1.x * wS0, rs1.y * wS1};
        const v2f t0 = {rt0.x * wT0, rt0.y * wT1};
        const v2f t1 = {rt1.x * wT0, rt1.y * wT1};

        // same registers serve as A (X_half rows) and B ((Y_half)^T cols)
        acc[0] = wmma4(s0, s0, acc[0]);  // SS tile (0,0)
        acc[1] = wmma4(s0, s1, acc[1]);  // SS tile (0,1)
        acc[2] = wmma4(s1, s1, acc[2]);  // SS tile (1,1)
        acc[3] = wmma4(t0, t0, acc[3]);  // TT tile (0,0)
        acc[4] = wmma4(t0, t1, acc[4]);  // TT tile (0,1)
        acc[5] = wmma4(t1, t1, acc[5]);  // TT tile (1,1)
        acc[6] = wmma4(s0, t0, acc[6]);  // ST tile (0,0)
        acc[7] = wmma4(s0, t1, acc[7]);  // ST tile (0,1)
        acc[8] = wmma4(s1, t0, acc[8]);  // ST tile (1,0)
        acc[9] = wmma4(s1, t1, acc[9]);  // ST tile (1,1)
    }

    // ---- spill tiles to LDS; C/D layout: VGPR r, lane -> row = r + 8*half, col = cl
    #pragma unroll
    for (int k = 0; k < TILES; ++k) {
        #pragma unroll
        for (int r = 0; r < 8; ++r) {
            const int row = r + 8 * half;
            gram[wave][k * TILE_ELEMS + row * 16 + cl] = acc[k][r];
        }
    }
    __syncthreads();

    // ---- reduce the 4 waves (fixed order -> deterministic), write partial Gram
    float* outp = wsGram + (size_t)(n * SPLIT + s) * GRAM_FLOATS;
    for (int i = t; i < GRAM_FLOATS; i += TPB) {
        outp[i] = ((gram[0][i] + gram[1][i]) + (gram[2][i] + gram[3][i]));
    }
}

// Kernel 2: sum the 16 M-split partials per map, Frobenius-weight, reduce to scalar.
__global__ void __launch_bounds__(256)
finalize_kernel(const float* __restrict__ wsGram, float* __restrict__ out) {
    __shared__ float red[256];
    const int t = threadIdx.x;
    // ||G32||^2 from 16x16 tiles: SS/TT use symmetry (1,2,1); ST counts -2 each.
    const float wgt[TILES] = {1.f, 2.f, 1.f, 1.f, 2.f, 1.f, -2.f, -2.f, -2.f, -2.f};

    float acc = 0.f;
    for (int n = 0; n < NMAT; ++n) {
        #pragma unroll
        for (int k = 0; k < TILES; ++k) {
            float g = 0.f;
            #pragma unroll
            for (int s = 0; s < SPLIT; ++s) {
                g += wsGram[(size_t)(n * SPLIT + s) * GRAM_FLOATS + k * TILE_ELEMS + t];
            }
            acc += wgt[k] * g * g;
        }
    }
    red[t] = acc;
    __syncthreads();
    for (int off = 128; off > 0; off >>= 1) {
        if (t < off) red[t] += red[t + off];
        __syncthreads();
    }
    if (t == 0) {
        // / (HW^2) / B / L  =  / (4096^2 * 8)
        out[0] = red[0] * (1.0f / (16777216.0f * 8.0f));
    }
}

extern "C" void kernel_launch(void* const* d_in, const int* in_sizes, int n_in,
                              void* d_out, int out_size, void* d_ws, size_t ws_size,
                              hipStream_t stream) {
    const float* S = (const float*)d_in[0];   // preds_S, fp32, 2*4*32*64*64
    const float* T = (const float*)d_in[1];   // preds_T
    float* ws  = (float*)d_ws;                // needs 128 * 2560 * 4B = 1.31 MB
    float* out = (float*)d_out;               // single fp32 scalar

    dim3 grid(SPLIT, NMAT);                   // 16 M-splits x 8 maps = 128 blocks
    gram_partial_kernel<<<grid, TPB, 0, stream>>>(S, T, ws);
    finalize_kernel<<<1, 256, 0, stream>>>(ws, out);
}